// MSTGCNBlock_71837622993319
// MI455X (gfx1250) — compile-verified
//
#include <hip/hip_runtime.h>

typedef __attribute__((ext_vector_type(2))) float v2f;
typedef __attribute__((ext_vector_type(8))) float v8f;

#define NNODES 10000
#define NBATCH 4
#define NTIME  8
#define NFIN   16
#define TBD    32            // T*B
#define CCHEB  32
#define CTIME  64
#define NROWS  320000        // TBD * NNODES
#define NRF    5120000       // NROWS * NFIN
#define LNEPS  1e-5f
#define PIters 64

// ---------------- utility ----------------
__global__ __launch_bounds__(256) void k_zero_int(int* __restrict__ p, int n) {
  int i = blockIdx.x * 256 + threadIdx.x;
  if (i < n) p[i] = 0;
}

// Build Xt[n][tb][f] replicating the torch-style permute/reshape scramble:
// Xt[tb,n2,f2] = Y[(n2*16+f2)*32+tb], Y = flat of transpose(X,(2,0,1,3)) i.e. (F,B,N,T)
__global__ __launch_bounds__(256) void k_build_xt(const float* __restrict__ X,
                                                  float* __restrict__ Xt) {
  int tid = blockIdx.x * 256 + threadIdx.x;          // < NRF
  int f2 = tid & 15;
  int tb = (tid >> 4) & 31;
  int n2 = tid >> 9;
  long i = ((long)(n2 * 16 + f2)) * 32 + tb;         // flat index into (F,B,N,T)
  int t = (int)(i & 7);
  long q = i >> 3;
  int n = (int)(q % NNODES);
  int p = (int)(q / NNODES);                         // p = f*4 + b
  int b = p & 3;
  int f = p >> 2;
  Xt[((long)n2 * 32 + tb) * 16 + f2] =
      X[(((long)(b * NNODES + n)) * 16 + f) * 8 + t];
}

// ---------------- CSR build (by dst) ----------------
__global__ __launch_bounds__(256) void k_count(const int* __restrict__ dst,
                                               int* __restrict__ indeg, int E) {
  int e = blockIdx.x * 256 + threadIdx.x;
  if (e < E) atomicAdd(&indeg[dst[e]], 1);
}

// single block exclusive scan: rowptr[0..NNODES] from indeg (chunk=10 per thread)
__global__ __launch_bounds__(1024) void k_scan(const int* __restrict__ indeg,
                                               int* __restrict__ rowptr) {
  __shared__ int part[1024];
  int tid = threadIdx.x;
  int base = tid * 10;
  int loc[10];
  int s = 0;
#pragma unroll
  for (int k = 0; k < 10; ++k) {
    int n = base + k;
    int d = (n < NNODES) ? indeg[n] : 0;
    loc[k] = d;
    s += d;
  }
  part[tid] = s;
  __syncthreads();
  for (int ofs = 1; ofs < 1024; ofs <<= 1) {
    int v = 0;
    if (tid >= ofs) v = part[tid - ofs];
    __syncthreads();
    part[tid] += v;
    __syncthreads();
  }
  int run = (tid == 0) ? 0 : part[tid - 1];
#pragma unroll
  for (int k = 0; k < 10; ++k) {
    int n = base + k;
    if (n < NNODES) rowptr[n] = run;
    run += loc[k];
  }
  if (tid == 1023) rowptr[NNODES] = run;  // == E
}

__global__ __launch_bounds__(256) void k_fill(const int* __restrict__ src,
                                              const int* __restrict__ dst,
                                              const int* __restrict__ rowptr,
                                              int* __restrict__ cursor,
                                              int* __restrict__ col, int E) {
  int e = blockIdx.x * 256 + threadIdx.x;
  if (e < E) {
    int d = dst[e];
    int pos = rowptr[d] + atomicAdd(&cursor[d], 1);
    col[pos] = src[e];
  }
}

// ---------------- fused 64-step power iteration (single WGP, CSR gather, LDS double-buffer)
// Dynamic LDS: va[10240] | vb[10240] | red[1024] = 86016 bytes (< 320KB/WGP)
__global__ __launch_bounds__(1024) void k_lambda(const int* __restrict__ rowptr,
                                                 const int* __restrict__ col,
                                                 float* __restrict__ sc) {
  extern __shared__ float sm[];
  float* va = sm;
  float* vb = sm + 10240;
  float* red = sm + 20480;
  const int tid = threadIdx.x;

  for (int i = tid; i < NNODES; i += 1024) va[i] = 1.0f / sqrtf((float)NNODES);
  __syncthreads();

  float nrm = 0.0f;
  float* cur = va;
  float* nxt = vb;
  for (int it = 0; it < PIters; ++it) {
    float ss = 0.0f;
    for (int i = tid; i < NNODES; i += 1024) {
      int beg = rowptr[i], end = rowptr[i + 1];
      float a = 0.0f;
      for (int k = beg; k < end; ++k) a += cur[col[k]];     // ds_load gather
      float w = (float)(end - beg) * cur[i] - a;            // (D - A) v
      nxt[i] = w;
      ss += w * w;
    }
    red[tid] = ss;
    __syncthreads();
    for (int s = 512; s > 0; s >>= 1) {
      if (tid < s) red[tid] += red[tid + s];
      __syncthreads();
    }
    nrm = sqrtf(red[0]);
    for (int i = tid; i < NNODES; i += 1024) nxt[i] /= nrm;
    __syncthreads();
    float* tmp = cur; cur = nxt; nxt = tmp;
  }
  if (tid == 0) sc[0] = nrm;   // lambda_max = last norm
}

// ---------------- Chebyshev recurrence: fused CSR gather + combine (atomic-free) --------
// Tx1[n,:] = diag(n)*Xt[n,:] - off * sum_{in-edges} Xt[src,:]
__global__ __launch_bounds__(256) void k_gather1(const int* __restrict__ rowptr,
                                                 const int* __restrict__ col,
                                                 const float* __restrict__ Xt,
                                                 const float* __restrict__ sc,
                                                 float* __restrict__ Tx1) {
  int tid = blockIdx.x * 256 + threadIdx.x;  // < NNODES*128
  int n = tid >> 7;
  int j = (tid & 127) << 2;
  int beg = rowptr[n], end = rowptr[n + 1];
  float4 acc = make_float4(0.f, 0.f, 0.f, 0.f);
  for (int k = beg; k < end; ++k) {
    float4 v = *(const float4*)(Xt + (long)col[k] * 512 + j);
    acc.x += v.x; acc.y += v.y; acc.z += v.z; acc.w += v.w;
  }
  float lm = sc[0];
  float off = 2.0f / lm;
  float dg = 2.0f * (float)(end - beg) / lm - 1.0f;
  float4 x0 = *(const float4*)(Xt + (long)n * 512 + j);
  *(float4*)(Tx1 + (long)n * 512 + j) =
      make_float4(dg * x0.x - off * acc.x, dg * x0.y - off * acc.y,
                  dg * x0.z - off * acc.z, dg * x0.w - off * acc.w);
}

// Tx2[n,:] = 2*(diag(n)*Tx1[n,:] - off * gather(Tx1)) - Xt[n,:]
__global__ __launch_bounds__(256) void k_gather2(const int* __restrict__ rowptr,
                                                 const int* __restrict__ col,
                                                 const float* __restrict__ Tx1,
                                                 const float* __restrict__ Xt,
                                                 const float* __restrict__ sc,
                                                 float* __restrict__ Tx2) {
  int tid = blockIdx.x * 256 + threadIdx.x;  // < NNODES*128
  int n = tid >> 7;
  int j = (tid & 127) << 2;
  int beg = rowptr[n], end = rowptr[n + 1];
  float4 acc = make_float4(0.f, 0.f, 0.f, 0.f);
  for (int k = beg; k < end; ++k) {
    float4 v = *(const float4*)(Tx1 + (long)col[k] * 512 + j);
    acc.x += v.x; acc.y += v.y; acc.z += v.z; acc.w += v.w;
  }
  float lm = sc[0];
  float off = 2.0f / lm;
  float dg = 2.0f * (float)(end - beg) / lm - 1.0f;
  float4 x1 = *(const float4*)(Tx1 + (long)n * 512 + j);
  float4 x0 = *(const float4*)(Xt + (long)n * 512 + j);
  *(float4*)(Tx2 + (long)n * 512 + j) =
      make_float4(2.0f * (dg * x1.x - off * acc.x) - x0.x,
                  2.0f * (dg * x1.y - off * acc.y) - x0.y,
                  2.0f * (dg * x1.z - off * acc.z) - x0.z,
                  2.0f * (dg * x1.w - off * acc.w) - x0.w);
}

// ---------------- Cheb GEMM: (320000x16)x(16x32) per k, WMMA f32 16x16x4 ----------------
__global__ __launch_bounds__(256) void k_cheb_gemm(const float* __restrict__ Tx0,
                                                   const float* __restrict__ Tx1,
                                                   const float* __restrict__ Tx2,
                                                   const float* __restrict__ W,
                                                   const float* __restrict__ bvec,
                                                   float* __restrict__ Out) {
  int wave = (blockIdx.x * 256 + threadIdx.x) >> 5;  // 0..19999 (one 16-row tile each)
  int lane = threadIdx.x & 31;
  int m = lane & 15;
  int half = lane >> 4;
  int kb = half * 2;
  long r0 = (long)wave * 16;
  long rowA = r0 + m;

  v8f c0 = {};
  v8f c1 = {};
  const float* Ts[3] = {Tx0, Tx1, Tx2};
#pragma unroll
  for (int kidx = 0; kidx < 3; ++kidx) {
    const float* Tk = Ts[kidx];
    const float* Wk = W + kidx * (NFIN * CCHEB);
#pragma unroll
    for (int kk = 0; kk < 16; kk += 4) {
      v2f a;
      a.x = Tk[rowA * 16 + kk + kb];
      a.y = Tk[rowA * 16 + kk + kb + 1];
      v2f b0, b1;
      b0.x = Wk[(kk + kb) * 32 + m];
      b0.y = Wk[(kk + kb + 1) * 32 + m];
      b1.x = Wk[(kk + kb) * 32 + 16 + m];
      b1.y = Wk[(kk + kb + 1) * 32 + 16 + m];
      c0 = __builtin_amdgcn_wmma_f32_16x16x4_f32(false, a, false, b0, (short)0, c0, false, false);
      c1 = __builtin_amdgcn_wmma_f32_16x16x4_f32(false, a, false, b1, (short)0, c1, false, false);
    }
  }
#pragma unroll
  for (int i = 0; i < 8; ++i) {
    long row = r0 + i + half * 8;
    float vA = c0[i] + bvec[m];
    float vB = c1[i] + bvec[16 + m];
    vA = vA > 0.0f ? vA : 0.0f;
    vB = vB > 0.0f ? vB : 0.0f;
    Out[row * 32 + m] = vA;
    Out[row * 32 + 16 + m] = vB;
  }
}

// ---------------- fused time-conv + residual-conv + bias + ReLU + LayerNorm ----------------
// rows r = n*32 + b*8 + t ; time conv K = 32*3 im2col ; residual K = 16 ; N = 64 channels
__global__ __launch_bounds__(256) void k_time_gemm(const float* __restrict__ Cheb,
                                                   const float* __restrict__ X,
                                                   const float* __restrict__ tW,
                                                   const float* __restrict__ tB,
                                                   const float* __restrict__ rW,
                                                   const float* __restrict__ rB,
                                                   const float* __restrict__ gamma,
                                                   const float* __restrict__ beta,
                                                   float* __restrict__ Out) {
  int wave = (blockIdx.x * 256 + threadIdx.x) >> 5;  // 0..19999
  int lane = threadIdx.x & 31;
  int m = lane & 15;
  int half = lane >> 4;
  int kb = half * 2;
  long r0 = (long)wave * 16;

  // A-row this lane feeds:
  long rA = r0 + m;
  int nA = (int)(rA >> 5);
  int tbA = (int)(rA & 31);
  int bA = tbA >> 3;
  int tA = tbA & 7;

  v8f acc0 = {}, acc1 = {}, acc2 = {}, acc3 = {};

  // ---- time conv: K = 96 (k = dt*32 + ci) ----
#pragma unroll
  for (int kk = 0; kk < 96; kk += 4) {
    int k0 = kk + kb;
    int dt0 = k0 >> 5, ci0 = k0 & 31;
    int dt1 = (k0 + 1) >> 5, ci1 = (k0 + 1) & 31;
    int tp0 = tA + dt0 - 1;
    int tp1 = tA + dt1 - 1;
    float m0 = (tp0 >= 0 && tp0 < 8) ? 1.0f : 0.0f;
    float m1 = (tp1 >= 0 && tp1 < 8) ? 1.0f : 0.0f;
    int tc0 = min(max(tp0, 0), 7);
    int tc1 = min(max(tp1, 0), 7);
    v2f a;
    a.x = m0 * Cheb[((long)nA * 32 + bA * 8 + tc0) * 32 + ci0];
    a.y = m1 * Cheb[((long)nA * 32 + bA * 8 + tc1) * 32 + ci1];
#define TCONV_STEP(ACC, CT)                                                        \
    {                                                                              \
      int co = (CT) * 16 + m;                                                      \
      v2f b;                                                                       \
      b.x = tW[co * 96 + ci0 * 3 + dt0];                                           \
      b.y = tW[co * 96 + ci1 * 3 + dt1];                                           \
      ACC = __builtin_amdgcn_wmma_f32_16x16x4_f32(false, a, false, b, (short)0,    \
                                                  ACC, false, false);              \
    }
    TCONV_STEP(acc0, 0)
    TCONV_STEP(acc1, 1)
    TCONV_STEP(acc2, 2)
    TCONV_STEP(acc3, 3)
#undef TCONV_STEP
  }

  // ---- residual conv: K = 16 (k = f) ----
#pragma unroll
  for (int kk = 0; kk < 16; kk += 4) {
    int k0 = kk + kb;
    v2f a;
    a.x = X[(((long)(bA * NNODES + nA)) * 16 + k0) * 8 + tA];
    a.y = X[(((long)(bA * NNODES + nA)) * 16 + k0 + 1) * 8 + tA];
#define RCONV_STEP(ACC, CT)                                                        \
    {                                                                              \
      int co = (CT) * 16 + m;                                                      \
      v2f b;                                                                       \
      b.x = rW[co * 16 + k0];                                                      \
      b.y = rW[co * 16 + k0 + 1];                                                  \
      ACC = __builtin_amdgcn_wmma_f32_16x16x4_f32(false, a, false, b, (short)0,    \
                                                  ACC, false, false);              \
    }
    RCONV_STEP(acc0, 0)
    RCONV_STEP(acc1, 1)
    RCONV_STEP(acc2, 2)
    RCONV_STEP(acc3, 3)
#undef RCONV_STEP
  }

  // ---- epilogue: bias + relu + LayerNorm(64) + scatter store ----
#pragma unroll
  for (int i = 0; i < 8; ++i) {
    long row = r0 + i + half * 8;
    int n = (int)(row >> 5);
    int tb = (int)(row & 31);
    int b = tb >> 3;
    int t = tb & 7;
    float vals[4];
    float s = 0.0f, ss = 0.0f;
    {
      float v0 = acc0[i] + tB[m] + rB[m];
      float v1 = acc1[i] + tB[16 + m] + rB[16 + m];
      float v2 = acc2[i] + tB[32 + m] + rB[32 + m];
      float v3 = acc3[i] + tB[48 + m] + rB[48 + m];
      v0 = v0 > 0.0f ? v0 : 0.0f;
      v1 = v1 > 0.0f ? v1 : 0.0f;
      v2 = v2 > 0.0f ? v2 : 0.0f;
      v3 = v3 > 0.0f ? v3 : 0.0f;
      vals[0] = v0; vals[1] = v1; vals[2] = v2; vals[3] = v3;
      s = v0 + v1 + v2 + v3;
      ss = v0 * v0 + v1 * v1 + v2 * v2 + v3 * v3;
    }
    // reduce across the 16 lanes of this half-wave (masks 1..8 stay within halves)
#pragma unroll
    for (int ofs = 1; ofs < 16; ofs <<= 1) {
      s += __shfl_xor(s, ofs, 32);
      ss += __shfl_xor(ss, ofs, 32);
    }
    float mu = s * (1.0f / 64.0f);
    float var = ss * (1.0f / 64.0f) - mu * mu;
    float inv = rsqrtf(var + LNEPS);
#pragma unroll
    for (int ct = 0; ct < 4; ++ct) {
      int c = ct * 16 + m;
      float y = (vals[ct] - mu) * inv * gamma[c] + beta[c];
      Out[(((long)(b * NNODES + n)) * 64 + c) * 8 + t] = y;
    }
  }
}

// ---------------- host launch ----------------
extern "C" void kernel_launch(void* const* d_in, const int* in_sizes, int n_in,
                              void* d_out, int out_size, void* d_ws, size_t ws_size,
                              hipStream_t stream) {
  (void)n_in; (void)out_size; (void)ws_size;
  const float* X     = (const float*)d_in[0];
  const int*   eidx  = (const int*)d_in[1];
  const float* chebW = (const float*)d_in[2];
  const float* chebB = (const float*)d_in[3];
  const float* timeW = (const float*)d_in[4];
  const float* timeB = (const float*)d_in[5];
  const float* resW  = (const float*)d_in[6];
  const float* resB  = (const float*)d_in[7];
  const float* gamma = (const float*)d_in[8];
  const float* beta  = (const float*)d_in[9];
  float* out = (float*)d_out;

  const int E = in_sizes[1] / 2;         // 160000
  const int* src = eidx;
  const int* dst = eidx + E;

  float* w    = (float*)d_ws;
  float* Xt   = w;                           // NRF
  float* Tx1  = Xt + NRF;                    // NRF
  float* Tx2  = Tx1 + NRF;                   // NRF
  float* Cheb = Tx2 + NRF;                   // NROWS*32
  float* sc   = Cheb + (size_t)NROWS * CCHEB; // sc[0] = lambda_max
  int*   ip     = (int*)(sc + 16);
  int*   indeg  = ip;                        // 10240
  int*   cursor = ip + 10240;                // 10240
  int*   rowptr = ip + 20480;                // 10496 (needs NNODES+1)
  int*   col    = ip + 30976;                // E

  // 1) build scrambled Xt (independent of graph work)
  k_build_xt<<<NRF / 256, 256, 0, stream>>>(X, Xt);

  // 2) CSR build: histogram -> scan -> fill
  k_zero_int<<<(20480 + 255) / 256, 256, 0, stream>>>(ip, 20480);  // indeg + cursor
  k_count<<<(E + 255) / 256, 256, 0, stream>>>(dst, indeg, E);
  k_scan<<<1, 1024, 0, stream>>>(indeg, rowptr);
  k_fill<<<(E + 255) / 256, 256, 0, stream>>>(src, dst, rowptr, cursor, col, E);

  // 3) fused 64-step power iteration on one WGP (LDS double-buffered vector)
  const size_t lambdaLds = (size_t)(2 * 10240 + 1024) * sizeof(float);  // 86016 B
  k_lambda<<<1, 1024, lambdaLds, stream>>>(rowptr, col, sc);

  // 4) Chebyshev recurrence: atomic-free fused gather+combine
  k_gather1<<<NNODES * 128 / 256, 256, 0, stream>>>(rowptr, col, Xt, sc, Tx1);
  k_gather2<<<NNODES * 128 / 256, 256, 0, stream>>>(rowptr, col, Tx1, Xt, sc, Tx2);

  // 5) cheb einsum GEMM (WMMA f32) + bias + relu
  k_cheb_gemm<<<2500, 256, 0, stream>>>(Xt, Tx1, Tx2, chebW, chebB, Cheb);

  // 6) fused time conv + residual conv + relu + LayerNorm (WMMA f32)
  k_time_gemm<<<2500, 256, 0, stream>>>(Cheb, X, timeW, timeB, resW, resB,
                                        gamma, beta, out);
}